// TopographLoss_72370198938230
// MI455X (gfx1250) — compile-verified
//
#include <hip/hip_runtime.h>
#include <hip/hip_bf16.h>
#include <stdint.h>

// Topograph loss for B=1, C=2 on a 64x112x112 volume.
// Pipeline: init -> union-find merge -> flatten -> region-graph min/max scan
//        -> critical mask + region counts -> n_reg -> WMMA block reduce -> final.

#define DD 64
#define HH 112
#define WW 112
#define NN (DD * HH * WW)   // 802816, divisible by 256

typedef float v2f __attribute__((ext_vector_type(2)));
typedef float v8f __attribute__((ext_vector_type(8)));

// ---------------- union-find helpers (lock-free min-tree) ----------------

__device__ __forceinline__ int uf_load(const int* p, int i) {
    return __hip_atomic_load(&p[i], __ATOMIC_RELAXED, __HIP_MEMORY_SCOPE_AGENT);
}

__device__ __forceinline__ int uf_find(const int* parent, int x) {
    int p = uf_load(parent, x);
    while (p != x) {            // parent[] is monotone non-increasing: terminates
        x = p;
        p = uf_load(parent, x);
    }
    return x;
}

__device__ __forceinline__ void uf_merge(int* parent, int u, int v) {
    int a = uf_find(parent, u);
    int b = uf_find(parent, v);
    while (a != b) {
        int hi = a > b ? a : b;
        int lo = a > b ? b : a;
        int old = atomicMin(&parent[hi], lo);
        if (old == hi) break;   // hi was a root, now linked to lo
        a = old;                // continue merging the previous link with lo
        b = lo;
    }
}

// ---------------- kernels ----------------

__global__ void k_init(const float* __restrict__ pred, const int* __restrict__ tgt,
                       int* __restrict__ parent, uint8_t* __restrict__ paired,
                       float* __restrict__ prob,
                       int* __restrict__ fgmin, int* __restrict__ fgmax,
                       int* __restrict__ bgmin, int* __restrict__ bgmax,
                       int* __restrict__ ccnt, int* __restrict__ nreg) {
    int i = blockIdx.x * blockDim.x + threadIdx.x;
    if (i >= NN) return;
    float x0 = pred[i];
    float x1 = pred[NN + i];
    int bp = (x1 > x0) ? 1 : 0;           // argmax over 2 channels, first-index tie break
    int bg = (tgt[i] == 1) ? 1 : 0;
    paired[i] = (uint8_t)(bp + 2 * bg);   // 0=TN 1=FP 2=FN 3=TP
    prob[i] = 1.0f / (1.0f + expf(x0 - x1));  // softmax channel 1 == sigmoid(x1-x0)
    parent[i] = i;
    fgmin[i] = 0x7fffffff; fgmax[i] = -1;
    bgmin[i] = 0x7fffffff; bgmax[i] = -1;
    ccnt[i] = 0;
    if (i == 0) *nreg = 0;
}

// Union with same-category neighbors; 26-conn for cats 1/2/3, 6-conn for cat 0.
// Only the 13 lexicographically-negative offsets (j < i): covers each pair once.
__global__ void k_merge(int* __restrict__ parent, const uint8_t* __restrict__ paired) {
    int i = blockIdx.x * blockDim.x + threadIdx.x;
    if (i >= NN) return;
    int x = i % WW;
    int y = (i / WW) % HH;
    int z = i / (WW * HH);
    uint8_t c = paired[i];
    bool face_only = (c == 0);
    #pragma unroll
    for (int dz = -1; dz <= 1; ++dz)
    #pragma unroll
    for (int dy = -1; dy <= 1; ++dy)
    #pragma unroll
    for (int dx = -1; dx <= 1; ++dx) {
        const bool neg_half = (dz < 0) || (dz == 0 && dy < 0) ||
                              (dz == 0 && dy == 0 && dx < 0);
        if (!neg_half) continue;
        const int nf = (dz != 0) + (dy != 0) + (dx != 0);
        int zz = z + dz, yy = y + dy, xx = x + dx;
        if ((unsigned)zz >= DD || (unsigned)yy >= HH || (unsigned)xx >= WW) continue;
        if (face_only && nf != 1) continue;
        int j = (zz * HH + yy) * WW + xx;
        if (paired[j] == c) uf_merge(parent, i, j);
    }
}

__global__ void k_flatten(int* __restrict__ parent) {
    int i = blockIdx.x * blockDim.x + threadIdx.x;
    if (i >= NN) return;
    int r = i;
    int p = uf_load(parent, r);
    while (p != r) { r = p; p = uf_load(parent, r); }
    parent[i] = r;   // label = min voxel index of component
}

// Region graph (26-conn, all-category dst). Only wrong-src components (FP/FN)
// ever feed the criticality test, so restrict src voxels to cats 1/2.
// "exactly one distinct neighbor comp" <=> exists && (min == max).
__global__ void k_edges(const int* __restrict__ labels, const uint8_t* __restrict__ paired,
                        int* __restrict__ fgmin, int* __restrict__ fgmax,
                        int* __restrict__ bgmin, int* __restrict__ bgmax) {
    int i = blockIdx.x * blockDim.x + threadIdx.x;
    if (i >= NN) return;
    uint8_t ci = paired[i];
    if (ci != 1 && ci != 2) return;       // only error components need neighbor stats
    int x = i % WW;
    int y = (i / WW) % HH;
    int z = i / (WW * HH);
    int li = labels[i];
    #pragma unroll
    for (int dz = -1; dz <= 1; ++dz)
    #pragma unroll
    for (int dy = -1; dy <= 1; ++dy)
    #pragma unroll
    for (int dx = -1; dx <= 1; ++dx) {
        if (dz == 0 && dy == 0 && dx == 0) continue;
        int zz = z + dz, yy = y + dy, xx = x + dx;
        if ((unsigned)zz >= DD || (unsigned)yy >= HH || (unsigned)xx >= WW) continue;
        int j = (zz * HH + yy) * WW + xx;
        int lj = labels[j];
        if (lj == li) continue;
        uint8_t cj = paired[j];
        if (cj == 3) {                     // TP neighbor component
            atomicMin(&fgmin[li], lj);
            atomicMax(&fgmax[li], lj);
        } else if (cj == 0) {              // TN neighbor component
            atomicMin(&bgmin[li], lj);
            atomicMax(&bgmax[li], lj);
        }
    }
}

__global__ void k_crit(const int* __restrict__ labels, const uint8_t* __restrict__ paired,
                       const int* __restrict__ fgmin, const int* __restrict__ fgmax,
                       const int* __restrict__ bgmin, const int* __restrict__ bgmax,
                       uint8_t* __restrict__ crit, int* __restrict__ ccnt) {
    int i = blockIdx.x * blockDim.x + threadIdx.x;
    if (i >= NN) return;
    uint8_t c = paired[i];
    bool wrong = (c == 1) || (c == 2);
    bool cr = false;
    if (wrong) {
        int l = labels[i];
        bool fg1 = (fgmax[l] >= 0) && (fgmin[l] == fgmax[l]);
        bool bg1 = (bgmax[l] >= 0) && (bgmin[l] == bgmax[l]);
        cr = !(fg1 && bg1);
        if (cr) atomicAdd(&ccnt[l], 1);    // integer atomics: order-independent
    }
    crit[i] = cr ? (uint8_t)1 : (uint8_t)0;
}

__global__ void k_nreg(const int* __restrict__ ccnt, int* __restrict__ nreg) {
    int i = blockIdx.x * blockDim.x + threadIdx.x;
    if (i >= NN) return;
    if (ccnt[i] > 0) atomicAdd(nreg, 1);
}

// Exact f32 sum of 256 shared-memory values with v_wmma_f32_16x16x4_f32:
// A = all-ones 16x4, B carries 64 data values per issue, C chains in f32.
// All D rows are identical; lanes 0..15 hold row 0 -> 4-step xor butterfly.
__device__ __forceinline__ float wmma_reduce_256(const float* sm) {
    int lane = threadIdx.x & 31;
    v8f c = {};
    v2f a; a.x = 1.0f; a.y = 1.0f;
    #pragma unroll
    for (int k = 0; k < 4; ++k) {
        v2f b;
        b.x = sm[k * 64 + lane * 2 + 0];
        b.y = sm[k * 64 + lane * 2 + 1];
        c = __builtin_amdgcn_wmma_f32_16x16x4_f32(false, a, false, b,
                                                  (short)0, c, false, false);
    }
    float v = c[0];
    v += __shfl_xor(v, 8, 32);
    v += __shfl_xor(v, 4, 32);
    v += __shfl_xor(v, 2, 32);
    v += __shfl_xor(v, 1, 32);
    return v;   // lanes 0..15 hold the total
}

// reg_sum = sum_i crit_i * (p_i - g_i)^2 / ccnt[label_i]
// (algebraically equal to sum over regions of per-region mean)
__global__ void k_loss_partial(const float* __restrict__ prob, const uint8_t* __restrict__ paired,
                               const uint8_t* __restrict__ crit, const int* __restrict__ labels,
                               const int* __restrict__ ccnt, float* __restrict__ partials) {
    __shared__ float sm[256];
    const int stride = gridDim.x * blockDim.x;
    float acc = 0.0f;
    for (int i = blockIdx.x * blockDim.x + threadIdx.x; i < NN; i += stride) {
        __builtin_prefetch(&crit[i + stride], 0, 0);   // global_prefetch_b8
        if (crit[i]) {
            float g = (float)((paired[i] >> 1) & 1);
            float d = prob[i] - g;
            acc += (d * d) / (float)ccnt[labels[i]];
        }
    }
    sm[threadIdx.x] = acc;
    __syncthreads();
    if (threadIdx.x < 32) {                 // wave-uniform branch: EXEC all-ones in wave 0
        float s = wmma_reduce_256(sm);
        if (threadIdx.x == 0) partials[blockIdx.x] = s;
    }
}

__global__ void k_final(const float* __restrict__ partials, const int* __restrict__ nreg,
                        float* __restrict__ out) {
    __shared__ float sm[256];
    sm[threadIdx.x] = partials[threadIdx.x];
    __syncthreads();
    if (threadIdx.x < 32) {
        float s = wmma_reduce_256(sm);
        if (threadIdx.x == 0) {
            int n = *nreg;
            out[0] = (n > 0) ? (s / (float)n) : 0.0f;
        }
    }
}

// ---------------- launch ----------------

extern "C" void kernel_launch(void* const* d_in, const int* in_sizes, int n_in,
                              void* d_out, int out_size, void* d_ws, size_t ws_size,
                              hipStream_t stream) {
    (void)in_sizes; (void)n_in; (void)out_size; (void)ws_size;
    const float* pred = (const float*)d_in[0];   // [1,2,64,112,112] f32
    const int*   tgt  = (const int*)d_in[1];     // [1,64,112,112] i32
    float* out = (float*)d_out;

    // workspace carve-up (256B aligned slices, ~24 MB total)
    char* ws = (char*)d_ws;
    size_t off = 0;
    auto carve = [&](size_t bytes) -> char* {
        char* p = ws + off;
        off += (bytes + 255) & ~(size_t)255;
        return p;
    };
    int*     parent   = (int*)carve((size_t)NN * 4);   // labels after flatten
    uint8_t* paired   = (uint8_t*)carve((size_t)NN);
    float*   prob     = (float*)carve((size_t)NN * 4);
    int*     fgmin    = (int*)carve((size_t)NN * 4);
    int*     fgmax    = (int*)carve((size_t)NN * 4);
    int*     bgmin    = (int*)carve((size_t)NN * 4);
    int*     bgmax    = (int*)carve((size_t)NN * 4);
    int*     ccnt     = (int*)carve((size_t)NN * 4);
    uint8_t* crit     = (uint8_t*)carve((size_t)NN);
    int*     nreg     = (int*)carve(4);
    float*   partials = (float*)carve(256 * 4);

    const dim3 blk(256);
    const int nblk = (NN + 255) / 256;   // 3136

    k_init<<<nblk, blk, 0, stream>>>(pred, tgt, parent, paired, prob,
                                     fgmin, fgmax, bgmin, bgmax, ccnt, nreg);
    k_merge<<<nblk, blk, 0, stream>>>(parent, paired);
    k_flatten<<<nblk, blk, 0, stream>>>(parent);
    k_edges<<<nblk, blk, 0, stream>>>(parent, paired, fgmin, fgmax, bgmin, bgmax);
    k_crit<<<nblk, blk, 0, stream>>>(parent, paired, fgmin, fgmax, bgmin, bgmax, crit, ccnt);
    k_nreg<<<nblk, blk, 0, stream>>>(ccnt, nreg);
    k_loss_partial<<<256, blk, 0, stream>>>(prob, paired, crit, parent, ccnt, partials);
    k_final<<<1, blk, 0, stream>>>(partials, nreg, out);
}